// EncoderGAE_23991687316148
// MI455X (gfx1250) — compile-verified
//
#include <hip/hip_runtime.h>
#include <hip/hip_bf16.h>

// ---------------------------------------------------------------------------
// GCN encoder (GAE) for MI455X / gfx1250.
//   N=50000 nodes, D=256, E=800000 edges, 8 GCN layers.
//   Dense 256x256 GEMMs use V_WMMA_F32_16X16X4_F32 (fp32-exact matrix path).
//   Edge aggregation: one wave32 per edge, global_atomic_add_f32 (hits L2;
//   50000x256 f32 feature matrix = 51MB fits in the 192MB L2).
// ---------------------------------------------------------------------------

#define N_NODES 50000
#define N_EDGES 800000
#define DIM     256
#define ROW_TILES (N_NODES / 16)   // 3125, exact
#define BN_EPS  1e-5f
#define SLOPE   0.01f

typedef __attribute__((ext_vector_type(2))) float v2f;
typedef __attribute__((ext_vector_type(8))) float v8f;

// ---------------- gcn_norm precompute ----------------

__global__ void k_deg_init(float* deg, int n) {
  int i = blockIdx.x * blockDim.x + threadIdx.x;
  if (i < n) deg[i] = 1.0f;  // self-loop weight
}

__global__ void k_deg_edges(const int* __restrict__ dst,
                            const float* __restrict__ ew,
                            float* __restrict__ deg, int e) {
  int i = blockIdx.x * blockDim.x + threadIdx.x;
  if (i < e) atomicAdd(&deg[dst[i]], ew[i]);
}

__global__ void k_dinv(float* deg_dinv, int n) {
  int i = blockIdx.x * blockDim.x + threadIdx.x;
  if (i < n) deg_dinv[i] = rsqrtf(deg_dinv[i]);  // deg >= 1 always
}

__global__ void k_norm(const int* __restrict__ src, const int* __restrict__ dst,
                       const float* __restrict__ ew, const float* __restrict__ dinv,
                       float* __restrict__ norm, int e) {
  int i = blockIdx.x * blockDim.x + threadIdx.x;
  if (i < e) norm[i] = dinv[src[i]] * ew[i] * dinv[dst[i]];
}

// ---------------- layer-1 GEMM: [N,6] x [6,256] ----------------

__global__ void k_gemm6(const float* __restrict__ x, const float* __restrict__ W,
                        float* __restrict__ y) {
  int idx = blockIdx.x * blockDim.x + threadIdx.x;
  if (idx >= N_NODES * DIM) return;
  int node = idx >> 8, d = idx & 255;
  const float* xr = x + (size_t)node * 6;
  float s = 0.0f;
#pragma unroll
  for (int k = 0; k < 6; ++k) s += xr[k] * W[k * DIM + d];
  y[idx] = s;
}

// ---------------- 256x256 GEMM via V_WMMA_F32_16X16X4_F32 ----------------
// One wave computes a 16-row x 64-col tile of y = x @ W.
// A fragment (16x4 f32): lane holds {x[m0+lane%16][k+2h], x[..][k+2h+1]}, h=lane/16.
// B fragment (4x16 f32): lane holds {W[k+2h][n+lane%16], W[k+2h+1][n+lane%16]}.
// C/D (16x16 f32): VGPR i -> row m0 + 8h + i, col = n + lane%16.
__global__ void k_gemm256_wmma(const float* __restrict__ x,
                               const float* __restrict__ W,
                               float* __restrict__ y) {
  const int lane = threadIdx.x & 31;
  const int waveId = (blockIdx.x * blockDim.x + threadIdx.x) >> 5;
  const int rowTile = waveId >> 2;          // /4 col strips
  const int colStrip = waveId & 3;
  if (rowTile >= ROW_TILES) return;
  const int m0 = rowTile * 16;
  const int n0 = colStrip * 64;
  const int r  = lane & 15;
  const int hk = (lane >> 4) * 2;           // k-pair offset: 0 or 2

  v8f acc0 = {}, acc1 = {}, acc2 = {}, acc3 = {};
  const float* xrow = x + (size_t)(m0 + r) * DIM;

  for (int k = 0; k < DIM; k += 4) {
    v2f a;
    a.x = xrow[k + hk];
    a.y = xrow[k + hk + 1];
    const float* wr0 = W + (size_t)(k + hk) * DIM + n0 + r;
    const float* wr1 = wr0 + DIM;
    v2f b0 = { wr0[0],  wr1[0]  };
    v2f b1 = { wr0[16], wr1[16] };
    v2f b2 = { wr0[32], wr1[32] };
    v2f b3 = { wr0[48], wr1[48] };
    acc0 = __builtin_amdgcn_wmma_f32_16x16x4_f32(false, a, false, b0, (short)0, acc0, false, false);
    acc1 = __builtin_amdgcn_wmma_f32_16x16x4_f32(false, a, false, b1, (short)0, acc1, false, false);
    acc2 = __builtin_amdgcn_wmma_f32_16x16x4_f32(false, a, false, b2, (short)0, acc2, false, false);
    acc3 = __builtin_amdgcn_wmma_f32_16x16x4_f32(false, a, false, b3, (short)0, acc3, false, false);
  }

  const int crow = m0 + (lane >> 4) * 8;
  const int c0   = n0 + r;
#pragma unroll
  for (int i = 0; i < 8; ++i) {
    float* yr = y + (size_t)(crow + i) * DIM + c0;
    yr[0]  = acc0[i];
    yr[16] = acc1[i];
    yr[32] = acc2[i];
    yr[48] = acc3[i];
  }
}

// ---------------- aggregation ----------------

// agg[n,d] = bias[d] + dinv[n]^2 * y[n,d]   (self-loop + bias, no atomics)
__global__ void k_agg_init(const float* __restrict__ y, const float* __restrict__ dinv,
                           const float* __restrict__ bias, float* __restrict__ agg) {
  int idx = blockIdx.x * blockDim.x + threadIdx.x;
  if (idx >= N_NODES * DIM) return;
  int node = idx >> 8, d = idx & 255;
  float di = dinv[node];
  agg[idx] = bias[d] + di * di * y[idx];
}

// one wave per edge: agg[dst,:] += norm * y[src,:]
__global__ void k_scatter(const float* __restrict__ y,
                          const int* __restrict__ src, const int* __restrict__ dst,
                          const float* __restrict__ norm, float* __restrict__ agg) {
  int wave = (blockIdx.x * blockDim.x + threadIdx.x) >> 5;
  int lane = threadIdx.x & 31;
  if (wave >= N_EDGES) return;
  int s = src[wave], t = dst[wave];
  float w = norm[wave];
  const float* ys = y + (size_t)s * DIM;
  float* at = agg + (size_t)t * DIM;
#pragma unroll
  for (int i = 0; i < DIM / 32; ++i)
    atomicAdd(&at[lane + 32 * i], w * ys[lane + 32 * i]);
}

// ---------------- batch norm + leaky relu ----------------

__global__ void k_zero_stats(float* stats) {
  int i = blockIdx.x * blockDim.x + threadIdx.x;
  if (i < 2 * DIM) stats[i] = 0.0f;
}

// per-column sum and sumsq; thread d owns column d within a row chunk
__global__ void k_bn_stats(const float* __restrict__ x, float* __restrict__ stats) {
  int d = threadIdx.x;  // 0..255
  int rows_per_block = (N_NODES + gridDim.x - 1) / gridDim.x;
  int r0 = blockIdx.x * rows_per_block;
  int r1 = r0 + rows_per_block;
  if (r1 > N_NODES) r1 = N_NODES;
  float s = 0.0f, s2 = 0.0f;
  for (int r = r0; r < r1; ++r) {
    float v = x[(size_t)r * DIM + d];
    s += v; s2 += v * v;
  }
  atomicAdd(&stats[d], s);
  atomicAdd(&stats[DIM + d], s2);
}

__global__ void k_bn_apply(const float* __restrict__ agg, const float* __restrict__ stats,
                           const float* __restrict__ gamma, const float* __restrict__ beta,
                           float* __restrict__ xo) {
  int idx = blockIdx.x * blockDim.x + threadIdx.x;
  if (idx >= N_NODES * DIM) return;
  int d = idx & 255;
  const float inv_n = 1.0f / (float)N_NODES;
  float mean = stats[d] * inv_n;
  float var  = stats[DIM + d] * inv_n - mean * mean;
  float sc   = gamma[d] * rsqrtf(var + BN_EPS);
  float v    = (agg[idx] - mean) * sc + beta[d];
  xo[idx] = v >= 0.0f ? v : SLOPE * v;
}

// ---------------- orchestration ----------------

extern "C" void kernel_launch(void* const* d_in, const int* in_sizes, int n_in,
                              void* d_out, int out_size, void* d_ws, size_t ws_size,
                              hipStream_t stream) {
  (void)in_sizes; (void)n_in; (void)out_size; (void)ws_size;

  const float* h   = (const float*)d_in[0];
  const int*   ei  = (const int*)d_in[1];
  const float* ew  = (const float*)d_in[2];
  const float* W1  = (const float*)d_in[3];
  const float* b1  = (const float*)d_in[4];
  const float* g1  = (const float*)d_in[5];
  const float* be1 = (const float*)d_in[6];
  const float* Wm  = (const float*)d_in[7];
  const float* bm  = (const float*)d_in[8];
  const float* gm  = (const float*)d_in[9];
  const float* bem = (const float*)d_in[10];
  const float* Wl  = (const float*)d_in[11];
  const float* bl  = (const float*)d_in[12];

  const int* src = ei;
  const int* dst = ei + N_EDGES;

  // workspace layout (floats)
  float* ws    = (float*)d_ws;
  float* dinv  = ws;                      size_t off = ((N_NODES + 255) / 256) * 256;
  float* norm  = ws + off;                off += N_EDGES;
  float* stats = ws + off;                off += 512;
  float* bufA  = ws + off;                off += (size_t)N_NODES * DIM;
  float* bufB  = ws + off;                off += (size_t)N_NODES * DIM;
  float* bufC  = ws + off;

  const int T = 256;
  const int gN    = (N_NODES + T - 1) / T;              // node-wise
  const int gE    = (N_EDGES + T - 1) / T;              // edge-wise
  const int gND   = (N_NODES * DIM) / T;                // elementwise [N,256]
  const int gScat = (N_EDGES * 32 + T - 1) / T;         // one wave per edge
  const int gGemm = (ROW_TILES * 4 + 7) / 8;            // 8 waves per 256-thr block

  // ---- gcn_norm ----
  k_deg_init<<<gN, T, 0, stream>>>(dinv, N_NODES);
  k_deg_edges<<<gE, T, 0, stream>>>(dst, ew, dinv, N_EDGES);
  k_dinv<<<gN, T, 0, stream>>>(dinv, N_NODES);
  k_norm<<<gE, T, 0, stream>>>(src, dst, ew, dinv, norm, N_EDGES);

  float* x = bufA;   // current activations
  float* y = bufB;   // gemm output
  float* agg = bufC; // aggregation output

  // ---- layer 1: 6 -> 256 ----
  k_gemm6<<<gND, T, 0, stream>>>(h, W1, y);
  k_agg_init<<<gND, T, 0, stream>>>(y, dinv, b1, agg);
  k_scatter<<<gScat, T, 0, stream>>>(y, src, dst, norm, agg);
  k_zero_stats<<<2, T, 0, stream>>>(stats);
  k_bn_stats<<<256, T, 0, stream>>>(agg, stats);
  k_bn_apply<<<gND, T, 0, stream>>>(agg, stats, g1, be1, x);

  // ---- middle layers: 6 x (256 -> 256), WMMA ----
  for (int i = 0; i < 6; ++i) {
    const float* Wi = Wm + (size_t)i * DIM * DIM;
    k_gemm256_wmma<<<gGemm, T, 0, stream>>>(x, Wi, y);
    k_agg_init<<<gND, T, 0, stream>>>(y, dinv, bm + i * DIM, agg);
    k_scatter<<<gScat, T, 0, stream>>>(y, src, dst, norm, agg);
    k_zero_stats<<<2, T, 0, stream>>>(stats);
    k_bn_stats<<<256, T, 0, stream>>>(agg, stats);
    k_bn_apply<<<gND, T, 0, stream>>>(agg, stats, gm + i * DIM, bem + i * DIM, x);
  }

  // ---- final layer: 256 -> 256, no BN/ReLU, straight into d_out ----
  float* out = (float*)d_out;
  k_gemm256_wmma<<<gGemm, T, 0, stream>>>(x, Wl, y);
  k_agg_init<<<gND, T, 0, stream>>>(y, dinv, bl, out);
  k_scatter<<<gScat, T, 0, stream>>>(y, src, dst, norm, out);
}